// EntityResolution_90305982366146
// MI455X (gfx1250) — compile-verified
//
#include <hip/hip_runtime.h>

typedef __attribute__((ext_vector_type(2))) float v2f;
typedef __attribute__((ext_vector_type(8))) float v8f;

#define WG 256

// ---------------------------------------------------------------- utilities
__global__ void zero_f32(float* __restrict__ p, long n) {
  long i = (long)blockIdx.x * blockDim.x + threadIdx.x;
  const long stride = (long)gridDim.x * blockDim.x;
  for (; i < n; i += stride) p[i] = 0.0f;
}

// per-destination in-degree for both edge types (same graph both layers)
__global__ void count_dst(const int* __restrict__ dA, const int* __restrict__ dB,
                          float* __restrict__ cA, float* __restrict__ cB, long E) {
  long i = (long)blockIdx.x * blockDim.x + threadIdx.x;
  const long stride = (long)gridDim.x * blockDim.x;
  for (; i < E; i += stride) {
    atomicAdd(&cA[dA[i]], 1.0f);
    atomicAdd(&cB[dB[i]], 1.0f);
  }
}

// copy_u + segment_sum: one thread per (edge, feature); L2-resident atomics
__global__ void scatter_h64(const float* __restrict__ msg, const int* __restrict__ src,
                            const int* __restrict__ dst, float* __restrict__ acc, long E) {
  const long total = E << 6;  // E * 64
  long i = (long)blockIdx.x * blockDim.x + threadIdx.x;
  const long stride = (long)gridDim.x * blockDim.x;
  for (; i < total; i += stride) {
    const long e = i >> 6;
    const int f = (int)(i & 63);
    const int s = src[e];
    const int d = dst[e];
    atomicAdd(&acc[(long)d * 64 + f], msg[(long)s * 64 + f]);
  }
}

// acc := acc / max(cnt,1), optional leaky_relu (alpha = 0.01, jax default)
__global__ void mean_act(float* __restrict__ acc, const float* __restrict__ cnt,
                         long nodes, int do_leaky) {
  const long total = nodes << 6;
  long i = (long)blockIdx.x * blockDim.x + threadIdx.x;
  const long stride = (long)gridDim.x * blockDim.x;
  for (; i < total; i += stride) {
    const long node = i >> 6;
    float c = cnt[node];
    c = (c < 1.0f) ? 1.0f : c;
    float v = acc[i] / c;
    if (do_leaky) v = (v >= 0.0f) ? v : 0.01f * v;
    acc[i] = v;
  }
}

// ------------------------------------------------------- WMMA fp32 GEMM N=64
// out[M,64] = concat(A1[M,k1], A2[M,k2]) @ B[K,64] + bias, K = k1+k2 <= 96.
// 256 threads = 8 waves; block tile = 32(M) x 64(N); wave tile = 16x16.
// V_WMMA_F32_16X16X4_F32 fragment layout (ISA 7.12.2):
//   A: lane l holds A[l&15][k0 + 2*(l>>4) + {0,1}]
//   B: lane l holds B[k0 + 2*(l>>4) + {0,1}][l&15]   (mirror of A)
//   C/D: vgpr i, lane l -> C[i + 8*(l>>4)][l&15]
__global__ void gemm_n64_wmma(const float* __restrict__ A1, int k1,
                              const float* __restrict__ A2, int k2,
                              const float* __restrict__ B,
                              const float* __restrict__ bias,
                              float* __restrict__ out, int M) {
  const int K = k1 + k2;          // 96 or 64
  __shared__ float sA[32 * 97];   // padded stride -> conflict-free column reads
  __shared__ float sB[96 * 64];

  const int tid = threadIdx.x;
  const int m0  = blockIdx.x * 32;

  // stage B [K x 64] (row-major), coalesced
  for (int i = tid; i < K * 64; i += WG) sB[i] = B[i];

  // stage A tile [32 x K] from the two concatenated sources, zero-pad tail rows
  for (int i = tid; i < 32 * K; i += WG) {
    const int r = i / K, c = i - r * K;
    const int m = m0 + r;
    float v = 0.0f;
    if (m < M) v = (c < k1) ? A1[(long)m * k1 + c] : A2[(long)m * k2 + (c - k1)];
    sA[r * 97 + c] = v;
  }
  __syncthreads();

  const int wave = tid >> 5;        // 0..7
  const int lane = tid & 31;
  const int mi   = wave >> 2;       // 0..1  (M sub-tile)
  const int ni   = wave & 3;        // 0..3  (N sub-tile)
  const int lm   = lane & 15;
  const int kk   = (lane >> 4) * 2; // 0 or 2

  const float* aRow = &sA[(mi * 16 + lm) * 97];
  const float* bCol = &sB[ni * 16 + lm];

  v8f c = {};
  for (int k0 = 0; k0 < K; k0 += 4) {
    v2f a, b;
    a.x = aRow[k0 + kk];
    a.y = aRow[k0 + kk + 1];
    b.x = bCol[(k0 + kk) * 64];
    b.y = bCol[(k0 + kk + 1) * 64];
    c = __builtin_amdgcn_wmma_f32_16x16x4_f32(
        /*neg_a=*/false, a, /*neg_b=*/false, b,
        /*c_mod=*/(short)0, c, /*reuse_a=*/false, /*reuse_b=*/false);
  }

  const int n  = ni * 16 + lm;
  const float bs = bias[n];
  const int mbase = m0 + mi * 16 + ((lane >> 4) ? 8 : 0);
#pragma unroll
  for (int i = 0; i < 8; ++i) {
    const int m = mbase + i;
    if (m < M) out[(long)m * 64 + n] = c[i] + bs;
  }
}

// ---------------------------------------------------------------- launcher
extern "C" void kernel_launch(void* const* d_in, const int* in_sizes, int n_in,
                              void* d_out, int out_size, void* d_ws, size_t ws_size,
                              hipStream_t stream) {
  const long NU = in_sizes[0];   // 200000
  const long NW = in_sizes[1];   // 50000
  const long E  = in_sizes[4];   // 1000000

  const float* user_features = (const float*)d_in[2];   // [NU,32]
  const int*   visits_src    = (const int*)d_in[4];
  const int*   visits_dst    = (const int*)d_in[5];
  const int*   vb_src        = (const int*)d_in[6];
  const int*   vb_dst        = (const int*)d_in[7];
  const float* user_emb      = (const float*)d_in[8];   // [NU,64]
  const float* Wv0           = (const float*)d_in[10];  // [96,64]
  const float* bv0           = (const float*)d_in[11];  // [64]
  const float* Wb1           = (const float*)d_in[16];  // [64,64]
  const float* bb1           = (const float*)d_in[17];  // [64]
  // dead per dependency analysis: website feats/emb, Wb0/bb0, Wv1/bv1

  float* ws       = (float*)d_ws;
  float* msg_u    = ws;                    // [NU,64] layer-0 user messages
  float* acc_web  = msg_u   + NU * 64;     // [NW,64] -> becomes h_web (activated)
  float* msg_w    = acc_web + NW * 64;     // [NW,64] layer-1 website messages
  float* cnt_web  = msg_w   + NW * 64;     // [NW]
  float* cnt_user = cnt_web + NW;          // [NU]
  float* out      = (float*)d_out;         // [NU,64] doubles as acc_user

  // zero accumulators + counters (contiguous ws region) and output
  zero_f32<<<4096, WG, 0, stream>>>(acc_web, NW * 64 * 2 + NW + NU);
  zero_f32<<<4096, WG, 0, stream>>>(out, NU * 64);

  // in-degrees (shared by both layers)
  count_dst<<<2048, WG, 0, stream>>>(visits_dst, vb_dst, cnt_web, cnt_user, E);

  // layer 0, user->web path only: msg_u = [user_emb | user_features] @ Wv0 + bv0
  gemm_n64_wmma<<<(int)((NU + 31) / 32), WG, 0, stream>>>(
      user_emb, 64, user_features, 32, Wv0, bv0, msg_u, (int)NU);

  // h_web accumulation over 'visits' edges
  scatter_h64<<<8192, WG, 0, stream>>>(msg_u, visits_src, visits_dst, acc_web, E);

  // h_web = leaky_relu(sum / max(cnt,1))   (in place)
  mean_act<<<2048, WG, 0, stream>>>(acc_web, cnt_web, NW, 1);

  // layer 1, web->user path only: msg_w = h_web @ Wb1 + bb1
  gemm_n64_wmma<<<(int)((NW + 31) / 32), WG, 0, stream>>>(
      acc_web, 64, nullptr, 0, Wb1, bb1, msg_w, (int)NW);

  // h_user accumulation over 'visited_by' edges, directly into d_out
  scatter_h64<<<8192, WG, 0, stream>>>(msg_w, vb_src, vb_dst, out, E);

  // final mean (no activation on the returned tensor)
  mean_act<<<4096, WG, 0, stream>>>(out, cnt_user, NU, 0);
}